// TATT_1_36034775613649
// MI455X (gfx1250) — compile-verified
//
#include <hip/hip_runtime.h>
#include <math.h>

#define B_  8
#define C_  256
#define N_  1024
#define T_  60
#define TP  64   // T padded to multiple of 16 for WMMA tiles

typedef __attribute__((ext_vector_type(2))) float v2f;
typedef __attribute__((ext_vector_type(8))) float v8f;

// ---------------------------------------------------------------------------
// Workspace layout (floats):
//   f1    [B][TP][N]   : 8*64*1024 = 524288
//   f2    [B][C][TP]   : 8*256*64  = 131072
//   g     [B][TP][C]   : 131072
//   sig   [B][TP][TP]  : 32768
//   vlog  [B][TP][TP]  : 32768
//   bn_sum[TP], bn_sq[TP]
// total = 852096 floats (~3.3 MB)
// ---------------------------------------------------------------------------
#define F1_ELEMS   ((size_t)B_ * TP * N_)
#define F2_ELEMS   ((size_t)B_ * C_ * TP)
#define G_ELEMS    ((size_t)B_ * TP * C_)
#define SIG_ELEMS  ((size_t)B_ * TP * TP)
#define WS_FLOATS  (F1_ELEMS + F2_ELEMS + G_ELEMS + 2 * SIG_ELEMS + 2 * TP)

// -------------------------- workspace zero-init ----------------------------
__global__ __launch_bounds__(256) void k_zero(float* __restrict__ ws, size_t n) {
    size_t i = (size_t)blockIdx.x * blockDim.x + threadIdx.x;
    if (i < n) ws[i] = 0.0f;
}

// ---------------------------------------------------------------------------
// Phase 1: single streaming pass over seq (503 MB -> bandwidth bound).
//   f1[b][t][n] += w1[c]*seq[b,c,n,t]   (register accum over c-quarter, atomic)
//   f2[b][c][t] += w2[n]*seq[b,c,n,t]   (register + shfl_xor reduce over 64 n)
// grid: 512 blocks = b(8) x n-slab(16) x c-quarter(4); 480 threads (15 waves)
// wave w owns t-pairs (4w..4w+3); lane: qh=l>>4 picks t-pair, nl=l&15 picks
// n within slab; each thread covers 4 n-rows (nl, nl+16, nl+32, nl+48).
// ---------------------------------------------------------------------------
__global__ __launch_bounds__(480) void k_phase1(const float* __restrict__ seq,
                                                const float* __restrict__ w1,
                                                const float* __restrict__ w2,
                                                float* __restrict__ f1,
                                                float* __restrict__ f2) {
    const int blk = blockIdx.x;
    const int b   = blk >> 6;          // 8 batches
    const int r   = blk & 63;
    const int nb  = r >> 2;            // 16 n-slabs of 64
    const int ch  = r & 3;             // 4 c-quarters of 64
    const int n0  = nb * 64;
    const int c0  = ch * 64;

    const int tid  = threadIdx.x;
    const int wave = tid >> 5;         // 0..14
    const int lane = tid & 31;
    const int qh   = lane >> 4;        // 0/1
    const int nl   = lane & 15;
    const int q    = wave * 2 + qh;    // 0..29 t-pair index
    const int t0   = q * 2;            // 0..58

    const float w2v0 = w2[n0 + nl];
    const float w2v1 = w2[n0 + nl + 16];
    const float w2v2 = w2[n0 + nl + 32];
    const float w2v3 = w2[n0 + nl + 48];

    v2f a0 = 0.f, a1 = 0.f, a2 = 0.f, a3 = 0.f;

    const size_t rowStride = (size_t)T_;       // floats between n rows
    const size_t cStride   = (size_t)N_ * T_;  // floats between c planes
    const float* p = seq + (((size_t)b * C_ + c0) * N_ + (size_t)(n0 + nl)) * T_ + t0;

    for (int c = 0; c < 64; ++c) {
        // 8B-aligned non-temporal streaming loads (seq is read exactly once)
        v2f x0 = __builtin_nontemporal_load((const v2f*)(p));
        v2f x1 = __builtin_nontemporal_load((const v2f*)(p + 16 * rowStride));
        v2f x2 = __builtin_nontemporal_load((const v2f*)(p + 32 * rowStride));
        v2f x3 = __builtin_nontemporal_load((const v2f*)(p + 48 * rowStride));

        const float wc = w1[c0 + c];           // uniform -> scalar load
        a0 += wc * x0; a1 += wc * x1; a2 += wc * x2; a3 += wc * x3;

        v2f pp = w2v0 * x0 + w2v1 * x1 + w2v2 * x2 + w2v3 * x3;
        float px = pp.x, py = pp.y;
        // reduce remaining 16-way n split across low 4 lane bits
        px += __shfl_xor(px, 1);  py += __shfl_xor(py, 1);
        px += __shfl_xor(px, 2);  py += __shfl_xor(py, 2);
        px += __shfl_xor(px, 4);  py += __shfl_xor(py, 4);
        px += __shfl_xor(px, 8);  py += __shfl_xor(py, 8);
        if (nl == 0) {
            float* dst = f2 + ((size_t)b * C_ + (size_t)(c0 + c)) * TP + t0;
            atomicAdd(dst + 0, px);
            atomicAdd(dst + 1, py);
        }
        p += cStride;
    }

    // f1 writeback: one add per element per c-quarter (4 contributors total)
    float* f1b = f1 + ((size_t)b * TP + t0) * N_ + (size_t)(n0 + nl);
    atomicAdd(f1b + 0,       a0.x); atomicAdd(f1b + N_ + 0,  a0.y);
    atomicAdd(f1b + 16,      a1.x); atomicAdd(f1b + N_ + 16, a1.y);
    atomicAdd(f1b + 32,      a2.x); atomicAdd(f1b + N_ + 32, a2.y);
    atomicAdd(f1b + 48,      a3.x); atomicAdd(f1b + N_ + 48, a3.y);
}

// ---------------------------------------------------------------------------
// WMMA helper: D = A(16x4 f32) * B(4x16 f32) + C, full fp32 precision.
// ---------------------------------------------------------------------------
__device__ __forceinline__ v8f wmma_f32(v2f a, v2f b, v8f c) {
    return __builtin_amdgcn_wmma_f32_16x16x4_f32(false, a, false, b,
                                                 (short)0, c, false, false);
}

// ---------------------------------------------------------------------------
// Phase 2: g[b][t][c] = sum_n f1[b][t][n] * w[n][c]
// one wave per 16x16 tile: 8 batches * 4 m-tiles * 16 c-tiles = 512 waves
// ---------------------------------------------------------------------------
__global__ __launch_bounds__(256) void k_gemm1(const float* __restrict__ f1,
                                               const float* __restrict__ w,
                                               float* __restrict__ g) {
    const int wid  = blockIdx.x * 8 + (threadIdx.x >> 5);   // 0..511
    const int lane = threadIdx.x & 31;
    const int b  = wid >> 6;
    const int rem = wid & 63;
    const int m0 = (rem >> 4) * 16;
    const int c0 = (rem & 15) * 16;
    const int h  = lane >> 4, lr = lane & 15;

    const float* ap = f1 + ((size_t)b * TP + (size_t)(m0 + lr)) * N_;  // A row (M=lr)
    const float* bp = w + (size_t)(c0 + lr);                           // B col (N=lr)
    v8f acc = {};
#pragma unroll 4
    for (int k = 0; k < N_; k += 4) {
        const int ka = k + 2 * h;                 // K = ka, ka+1 for this half
        v2f av = *(const v2f*)(ap + ka);
        v2f bv;
        bv.x = bp[(size_t)ka * C_];
        bv.y = bp[(size_t)(ka + 1) * C_];
        acc = wmma_f32(av, bv, acc);
    }
    float* gp = g + ((size_t)b * TP + (size_t)(m0 + 8 * h)) * C_ + (size_t)(c0 + lr);
#pragma unroll
    for (int rr = 0; rr < 8; ++rr) gp[(size_t)rr * C_] = acc[rr];
}

// ---------------------------------------------------------------------------
// Phase 3: sig[b][t][s] = sigmoid( sum_c g[b][t][c]*f2[b][c][s] + bias[t][s] )
// 8 batches * 4 m-tiles * 4 s-tiles = 128 waves; pad rows/cols stored as 0.
// ---------------------------------------------------------------------------
__global__ __launch_bounds__(256) void k_gemm2(const float* __restrict__ g,
                                               const float* __restrict__ f2,
                                               const float* __restrict__ bias,
                                               float* __restrict__ sig) {
    const int wid  = blockIdx.x * 8 + (threadIdx.x >> 5);   // 0..127
    const int lane = threadIdx.x & 31;
    const int b  = wid >> 4;
    const int rem = wid & 15;
    const int m0 = (rem >> 2) * 16;
    const int s0 = (rem & 3) * 16;
    const int h  = lane >> 4, lr = lane & 15;

    const float* ap = g + ((size_t)b * TP + (size_t)(m0 + lr)) * C_;
    const float* bp = f2 + (size_t)b * C_ * TP + (size_t)(s0 + lr);
    v8f acc = {};
#pragma unroll 4
    for (int k = 0; k < C_; k += 4) {
        const int ka = k + 2 * h;
        v2f av = *(const v2f*)(ap + ka);
        v2f bv;
        bv.x = bp[(size_t)ka * TP];
        bv.y = bp[(size_t)(ka + 1) * TP];
        acc = wmma_f32(av, bv, acc);
    }
    const int s = s0 + lr;
#pragma unroll
    for (int rr = 0; rr < 8; ++rr) {
        const int t = m0 + 8 * h + rr;
        float val = 0.0f;
        if (t < T_ && s < T_) {
            const float z = acc[rr] + bias[t * T_ + s];
            val = 1.0f / (1.0f + __expf(-z));
        }
        sig[((size_t)b * TP + t) * TP + s] = val;   // pads written as exact 0
    }
}

// ---------------------------------------------------------------------------
// Phase 4: vlog[b][i][s] = sum_j v[i][j]*sig[b][j][s]; accumulate BN stats
// (sum and sum-of-squares per column s over all b and rows i<60).
// ---------------------------------------------------------------------------
__global__ __launch_bounds__(256) void k_gemm3(const float* __restrict__ vm,
                                               const float* __restrict__ sig,
                                               float* __restrict__ vlog,
                                               float* __restrict__ bn_sum,
                                               float* __restrict__ bn_sq) {
    const int wid  = blockIdx.x * 8 + (threadIdx.x >> 5);   // 0..127
    const int lane = threadIdx.x & 31;
    const int b  = wid >> 4;
    const int rem = wid & 15;
    const int m0 = (rem >> 2) * 16;
    const int s0 = (rem & 3) * 16;
    const int h  = lane >> 4, lr = lane & 15;

    const int i_row = m0 + lr;                               // A-matrix M index
    const float* bp = sig + (size_t)b * TP * TP + (size_t)(s0 + lr);
    v8f acc = {};
#pragma unroll
    for (int k = 0; k < TP; k += 4) {
        const int ka = k + 2 * h;
        v2f av;
        av.x = (i_row < T_ && (ka)     < T_) ? vm[i_row * T_ + ka]     : 0.0f;
        av.y = (i_row < T_ && (ka + 1) < T_) ? vm[i_row * T_ + ka + 1] : 0.0f;
        v2f bv;
        bv.x = bp[(size_t)ka * TP];          // sig pad rows are zero
        bv.y = bp[(size_t)(ka + 1) * TP];
        acc = wmma_f32(av, bv, acc);
    }
    const int s = s0 + lr;
    float ls = 0.0f, lsq = 0.0f;
#pragma unroll
    for (int rr = 0; rr < 8; ++rr) {
        const int i = m0 + 8 * h + rr;
        const float val = acc[rr];
        vlog[((size_t)b * TP + i) * TP + s] = val;
        if (i < T_) { ls += val; lsq += val * val; }
    }
    atomicAdd(&bn_sum[s], ls);
    atomicAdd(&bn_sq[s], lsq);
}

// ---------------------------------------------------------------------------
// Phase 5: per-row BatchNorm + block-diagonal mask + softmax.
// One wave32 per (b, i) row; lane covers s=lane and s=lane+32.
// ---------------------------------------------------------------------------
__global__ __launch_bounds__(256) void k_softmax(const float* __restrict__ vlog,
                                                 const float* __restrict__ bn_sum,
                                                 const float* __restrict__ bn_sq,
                                                 const float* __restrict__ gamma,
                                                 const float* __restrict__ beta,
                                                 float* __restrict__ out) {
    const int wid  = blockIdx.x * 8 + (threadIdx.x >> 5);   // 0..479
    const int lane = threadIdx.x & 31;
    const int b = wid / T_;
    const int i = wid % T_;
    const float inv_cnt = 1.0f / (float)(B_ * T_);
    const float* row = vlog + ((size_t)b * TP + i) * TP;
    const int bi = (i < 36) ? (i / 12) : 3;

    float y0, y1 = -INFINITY;
    {
        const int s = lane;                                  // always < 60
        const float mean = bn_sum[s] * inv_cnt;
        const float var  = bn_sq[s] * inv_cnt - mean * mean;
        float x = (row[s] - mean) * rsqrtf(var + 1e-5f) * gamma[s] + beta[s];
        const int bs = (s < 36) ? (s / 12) : 3;
        y0 = x + ((bs == bi) ? 0.0f : -1e13f);
    }
    if (lane + 32 < T_) {
        const int s = lane + 32;
        const float mean = bn_sum[s] * inv_cnt;
        const float var  = bn_sq[s] * inv_cnt - mean * mean;
        float x = (row[s] - mean) * rsqrtf(var + 1e-5f) * gamma[s] + beta[s];
        const int bs = (s < 36) ? (s / 12) : 3;
        y1 = x + ((bs == bi) ? 0.0f : -1e13f);
    }
    float m = fmaxf(y0, y1);
#pragma unroll
    for (int d = 16; d >= 1; d >>= 1) m = fmaxf(m, __shfl_xor(m, d));
    const float e0 = __expf(y0 - m);
    const float e1 = (lane + 32 < T_) ? __expf(y1 - m) : 0.0f;
    float ssum = e0 + e1;
#pragma unroll
    for (int d = 16; d >= 1; d >>= 1) ssum += __shfl_xor(ssum, d);
    const float inv = 1.0f / ssum;
    float* orow = out + ((size_t)b * T_ + i) * T_;
    orow[lane] = e0 * inv;
    if (lane + 32 < T_) orow[lane + 32] = e1 * inv;
}

// ---------------------------------------------------------------------------
extern "C" void kernel_launch(void* const* d_in, const int* in_sizes, int n_in,
                              void* d_out, int out_size, void* d_ws, size_t ws_size,
                              hipStream_t stream) {
    (void)in_sizes; (void)n_in; (void)out_size; (void)ws_size;
    const float* seq   = (const float*)d_in[0];
    const float* w1    = (const float*)d_in[1];
    const float* w2    = (const float*)d_in[2];
    const float* w     = (const float*)d_in[3];
    const float* bias  = (const float*)d_in[4];
    const float* vm    = (const float*)d_in[5];
    const float* gamma = (const float*)d_in[6];
    const float* beta  = (const float*)d_in[7];

    float* ws     = (float*)d_ws;
    float* f1     = ws;
    float* f2     = f1 + F1_ELEMS;
    float* g      = f2 + F2_ELEMS;
    float* sig    = g + G_ELEMS;
    float* vlog   = sig + SIG_ELEMS;
    float* bn_sum = vlog + SIG_ELEMS;
    float* bn_sq  = bn_sum + TP;

    // zero scratch (f1/f2 accumulators, pad rows, BN stats) — deterministic
    {
        const size_t n = WS_FLOATS;
        const int blocks = (int)((n + 255) / 256);
        k_zero<<<blocks, 256, 0, stream>>>(ws, n);
    }
    // phase 1: the single bandwidth-bound pass over seq
    k_phase1<<<512, 480, 0, stream>>>(seq, w1, w2, f1, f2);
    // phases 2-4: fp32 WMMA GEMM chain
    k_gemm1<<<64, 256, 0, stream>>>(f1, w, g);
    k_gemm2<<<16, 256, 0, stream>>>(g, f2, bias, sig);
    k_gemm3<<<16, 256, 0, stream>>>(vm, sig, vlog, bn_sum, bn_sq);
    // phase 5: BN + mask + softmax
    k_softmax<<<60, 256, 0, stream>>>(vlog, bn_sum, bn_sq, gamma, beta, (float*)d_out);
}